// OptimizedCmxuLayer_22376779612867
// MI455X (gfx1250) — compile-verified
//
#include <hip/hip_runtime.h>
#include <math.h>

#define N_DIM 128

typedef float v2f __attribute__((ext_vector_type(2)));
typedef float v8f __attribute__((ext_vector_type(8)));

// ---------------------------------------------------------------------------
// Kernel 1: build the 128x128 unitary from the MZI mesh using the 2D-wavefront
// (anti-diagonal) schedule. Each block owns 16 columns of U (columns evolve
// independently); threads = 16 columns x 64 diagonal slots; U columns in LDS.
// At the end, each thread writes exactly one float4 WMMA B-fragment entry:
//   upk[(kc*8 + tt)*32 + lane] = { ReU^T[k][n], ReU^T[k+1][n],
//                                  ImU^T[k][n], ImU^T[k+1][n] }
// with k = kc*4 + 2*(lane>>4), n = tt*16 + (lane&15), output phases folded in.
// One global_load_b128 in the GEMM then yields a full complex B fragment.
// ---------------------------------------------------------------------------
__global__ __launch_bounds__(1024) void build_unitary_kernel(
    const float* __restrict__ mzi,      // 8128*2 floats: (theta, phi) pairs
    const float* __restrict__ outph,    // 128 floats
    float4* __restrict__ upk)           // 32*8*32 float4 = 128 KB
{
    __shared__ float2 u[N_DIM * 16];    // u[row*16 + col_local]

    const int t    = threadIdx.x;
    const int cl   = t & 15;            // column within block: 0..15
    const int slot = t >> 4;            // diagonal slot: 0..63
    const int col  = blockIdx.x * 16 + cl;

    // init: column `col` of identity
    for (int n = slot; n < N_DIM; n += 64) {
        u[n * 16 + cl] = make_float2((n == col) ? 1.0f : 0.0f, 0.0f);
    }
    __syncthreads();

    const float Cc = 0.70710678118654752f;  // sqrt(0.5)

    // anti-diagonals d = i + j, d = 1 .. 253; steps on one diagonal touch
    // disjoint row pairs (independent), and diagonal order is a valid
    // linearization of the triu_indices dependency partial order.
    for (int d = 1; d <= 2 * N_DIM - 3; ++d) {
        const int imin = (d > N_DIM - 1) ? (d - (N_DIM - 1)) : 0;
        const int imax = (d - 1) >> 1;
        const int i = imin + slot;
        if (i <= imax) {
            const int j = d - i;
            // triu_indices(128,1) linear index of (i,j)
            const int idx = i * (N_DIM - 1) - (i * (i - 1)) / 2 + (j - i - 1);
            const float th = mzi[2 * idx];
            const float ph = mzi[2 * idx + 1];
            float sth, cth, sph, cph;
            __sincosf(th, &sth, &cth);
            __sincosf(ph, &sph, &cph);

            const float2 ri = u[i * 16 + cl];
            const float2 rj = u[j * 16 + cl];
            // tt = e^{i*phi} * rj
            const float tr = cph * rj.x - sph * rj.y;
            const float ti = cph * rj.y + sph * rj.x;
            // new_i = C*(ri + tt)
            float2 ni;
            ni.x = Cc * (ri.x + tr);
            ni.y = Cc * (ri.y + ti);
            // new_j = C * e^{i*theta} * (ri - tt)
            const float wr = ri.x - tr;
            const float wi = ri.y - ti;
            float2 nj;
            nj.x = Cc * (cth * wr - sth * wi);
            nj.y = Cc * (cth * wi + sth * wr);
            u[i * 16 + cl] = ni;
            u[j * 16 + cl] = nj;
        }
        __syncthreads();
    }
    // (loop ends with a barrier: all columns final)

    // Each of the 1024 threads writes one packed B-fragment float4 for the
    // 4 K-chunks owned by this block (16 k-rows = 16 columns of U we hold).
    const int kcl = t >> 8;                 // local K-chunk: 0..3
    const int tt  = (t >> 5) & 7;           // n-tile: 0..7
    const int ln  = t & 31;                 // lane: 0..31
    const int kh  = ln >> 4;                // K half-select
    const int nc  = tt * 16 + (ln & 15);    // output row n
    const int kr0 = kcl * 4 + 2 * kh;       // local k (even), 0..14
    const int kr1 = kr0 + 1;

    float s, c;
    __sincosf(outph[nc], &s, &c);           // row-n output phase
    const float2 u0 = u[nc * 16 + kr0];
    const float2 u1 = u[nc * 16 + kr1];
    float4 f;
    f.x = c * u0.x - s * u0.y;              // Re U[n][k]
    f.y = c * u1.x - s * u1.y;              // Re U[n][k+1]
    f.z = c * u0.y + s * u0.x;              // Im U[n][k]
    f.w = c * u1.y + s * u1.x;              // Im U[n][k+1]

    const int kc = blockIdx.x * 4 + kcl;    // global K-chunk 0..31
    upk[(kc * 8 + tt) * 32 + ln] = f;
}

// ---------------------------------------------------------------------------
// Kernel 2: complex GEMM out = x @ U^T with fp32 WMMA (V_WMMA_F32_16X16X4_F32).
// Each wave: 32 batch rows (two 16-row A tiles sharing B) x all 128 outputs
// (8 n-tiles). Per K-chunk: 4 b64 A loads + 8 b128 B loads -> 64 WMMAs.
//   out_r = xr*UrT + xi*(-UiT),   out_i = xr*UiT + xi*UrT
// U-pack (128 KB) is L2/L0 resident across all blocks.
// ---------------------------------------------------------------------------
__global__ __launch_bounds__(256) void cgemm_wmma_kernel(
    const float* __restrict__ xr,
    const float* __restrict__ xi,
    const float4* __restrict__ upk,
    float2* __restrict__ out)
{
    const int lane = threadIdx.x & 31;
    const int wave = threadIdx.x >> 5;
    const int b0   = blockIdx.x * 256 + wave * 32;

    const int mrow = lane & 15;   // A row within tile
    const int kh   = lane >> 4;   // K half-select

    v8f accr[2][8], acci[2][8];
#pragma unroll
    for (int g = 0; g < 2; ++g)
#pragma unroll
        for (int tt = 0; tt < 8; ++tt)
#pragma unroll
            for (int q = 0; q < 8; ++q) {
                accr[g][tt][q] = 0.0f;
                acci[g][tt][q] = 0.0f;
            }

    const float* xr0 = xr + (size_t)(b0 + mrow) * N_DIM + kh * 2;
    const float* xi0 = xi + (size_t)(b0 + mrow) * N_DIM + kh * 2;
    const float* xr1 = xr0 + 16 * N_DIM;
    const float* xi1 = xi0 + 16 * N_DIM;
    const float4* up = upk + lane;          // + imm offsets below

    for (int kc = 0; kc < 32; ++kc) {
        const int k0 = kc * 4;
        // A fragments (16x4 f32): vgpr0 = K(k0+2kh), vgpr1 = K(k0+2kh+1)
        const v2f ar0 = *(const v2f*)(xr0 + k0);
        const v2f ai0 = *(const v2f*)(xi0 + k0);
        const v2f ar1 = *(const v2f*)(xr1 + k0);
        const v2f ai1 = *(const v2f*)(xi1 + k0);

#pragma unroll
        for (int tt = 0; tt < 8; ++tt) {
            const float4 f = up[(kc * 8 + tt) * 32];
            v2f br, bi, bn;
            br.x = f.x;  br.y = f.y;        // Re fragment
            bi.x = f.z;  bi.y = f.w;        // Im fragment
            bn.x = -f.z; bn.y = -f.w;       // -Im fragment (2 VALU, no load)

            accr[0][tt] = __builtin_amdgcn_wmma_f32_16x16x4_f32(
                false, ar0, false, br, (short)0, accr[0][tt], false, false);
            accr[0][tt] = __builtin_amdgcn_wmma_f32_16x16x4_f32(
                false, ai0, false, bn, (short)0, accr[0][tt], false, false);
            acci[0][tt] = __builtin_amdgcn_wmma_f32_16x16x4_f32(
                false, ar0, false, bi, (short)0, acci[0][tt], false, false);
            acci[0][tt] = __builtin_amdgcn_wmma_f32_16x16x4_f32(
                false, ai0, false, br, (short)0, acci[0][tt], false, false);

            accr[1][tt] = __builtin_amdgcn_wmma_f32_16x16x4_f32(
                false, ar1, false, br, (short)0, accr[1][tt], false, false);
            accr[1][tt] = __builtin_amdgcn_wmma_f32_16x16x4_f32(
                false, ai1, false, bn, (short)0, accr[1][tt], false, false);
            acci[1][tt] = __builtin_amdgcn_wmma_f32_16x16x4_f32(
                false, ar1, false, bi, (short)0, acci[1][tt], false, false);
            acci[1][tt] = __builtin_amdgcn_wmma_f32_16x16x4_f32(
                false, ai1, false, br, (short)0, acci[1][tt], false, false);
        }
    }

    // C/D layout: vgpr v, lanes0-15 -> M=v, lanes16-31 -> M=v+8; N = lane%16
#pragma unroll
    for (int g = 0; g < 2; ++g) {
#pragma unroll
        for (int tt = 0; tt < 8; ++tt) {
#pragma unroll
            for (int v = 0; v < 8; ++v) {
                const int b = b0 + g * 16 + v + 8 * kh;
                float2 o;
                o.x = accr[g][tt][v];
                o.y = acci[g][tt][v];
                out[(size_t)b * N_DIM + tt * 16 + mrow] = o;
            }
        }
    }
}

extern "C" void kernel_launch(void* const* d_in, const int* in_sizes, int n_in,
                              void* d_out, int out_size, void* d_ws, size_t ws_size,
                              hipStream_t stream) {
    const float* xr  = (const float*)d_in[0];  // (B,128) f32
    const float* xi  = (const float*)d_in[1];  // (B,128) f32
    const float* mzi = (const float*)d_in[2];  // 16256 f32
    const float* oph = (const float*)d_in[3];  // 128 f32

    float4* upk = (float4*)d_ws;               // 8192 float4 = 128 KB

    // Stage 1: build packed U fragments. 8 blocks x 1024 threads.
    build_unitary_kernel<<<8, 1024, 0, stream>>>(mzi, oph, upk);

    // Stage 2: batched complex GEMM. B = 131072 rows / 256 per block.
    const int batch = in_sizes[0] / N_DIM;
    cgemm_wmma_kernel<<<batch / 256, 256, 0, stream>>>(
        xr, xi, upk, (float2*)d_out);
}